// BranchTeacherLayoutLoss_37074157699123
// MI455X (gfx1250) — compile-verified
//
#include <hip/hip_runtime.h>
#include <math.h>

#define D 256
#define EPS_BALL 1e-5f

typedef float v2f __attribute__((ext_vector_type(2)));
typedef float v8f __attribute__((ext_vector_type(8)));

// ---------------------------------------------------------------- zero
__global__ void zero_kernel(int* counts, float* out, int B, int out_n) {
  int t = blockIdx.x * blockDim.x + threadIdx.x;
  if (t < B) counts[t] = 0;
  if (t < out_n) out[t] = 0.0f;
}

// ---------------------------------------------------------------- counts
__global__ void count_kernel(const int* __restrict__ segids, int* counts, int M) {
  int t = blockIdx.x * blockDim.x + threadIdx.x;
  if (t < M) atomicAdd(&counts[segids[t]], 1);
}

// ---------------------------------------------------------------- exclusive scan (B <= 1024)
__global__ void scan_kernel(const int* __restrict__ counts, int* __restrict__ starts, int B) {
  __shared__ int buf[2][1024];
  int t = threadIdx.x;
  int cur = 0;
  if (t < B) buf[0][t] = counts[t];
  __syncthreads();
  for (int off = 1; off < B; off <<= 1) {
    int nxt = cur ^ 1;
    if (t < B) {
      int v = buf[cur][t];
      if (t >= off) v += buf[cur][t - off];
      buf[nxt][t] = v;
    }
    __syncthreads();
    cur = nxt;
  }
  if (t < B) starts[t] = buf[cur][t] - counts[t];
}

// ---------------------------------------------------------------- main gather + segment mean
// One block per segment; one wave per member; b128 loads; no barriers in hot loop.
__global__ __launch_bounds__(256)
void segment_kernel(const float* __restrict__ emb,
                    const int* __restrict__ midx,
                    const int* __restrict__ starts,
                    const int* __restrict__ counts,
                    const float* __restrict__ tcoh,
                    float* __restrict__ centroid_ws,
                    float* __restrict__ cohl_ws) {
  __shared__ float lacc[8][D];
  __shared__ float sred[16];

  const int b     = blockIdx.x;
  const int start = starts[b];
  const int cnt   = counts[b];
  const int wave  = threadIdx.x >> 5;
  const int lane  = threadIdx.x & 31;
  const float maxnorm = 1.0f - EPS_BALL;

  float acc[8];
#pragma unroll
  for (int j = 0; j < 8; ++j) acc[j] = 0.0f;

  const int end = start + cnt;
  for (int i = start + wave; i < end; i += 8) {
    const int idx = midx[i];
    const float4* row = (const float4*)(emb + (size_t)idx * D);
    float4 v0 = row[lane];        // dims 4*lane .. 4*lane+3
    float4 v1 = row[32 + lane];   // dims 128+4*lane ..

    // software prefetch of next member's row (global_prefetch_b8)
    if (i + 8 < end) {
      const int nidx = midx[i + 8];
      __builtin_prefetch(emb + (size_t)nidx * D + lane * 8, 0, 1);
    }

    float ss = v0.x * v0.x + v0.y * v0.y + v0.z * v0.z + v0.w * v0.w +
               v1.x * v1.x + v1.y * v1.y + v1.z * v1.z + v1.w * v1.w;
#pragma unroll
    for (int o = 16; o >= 1; o >>= 1) ss += __shfl_xor(ss, o, 32);

    // faithful to reference: project_to_ball then divide by max(norm,1e-8)
    const float nrm   = sqrtf(ss);
    const float scale = (nrm > maxnorm) ? (maxnorm / fmaxf(nrm, 1e-12f)) : 1.0f;
    const float pn    = nrm * scale;
    const float inv   = scale / fmaxf(pn, 1e-8f);

    acc[0] += v0.x * inv; acc[1] += v0.y * inv; acc[2] += v0.z * inv; acc[3] += v0.w * inv;
    acc[4] += v1.x * inv; acc[5] += v1.y * inv; acc[6] += v1.z * inv; acc[7] += v1.w * inv;
  }

#pragma unroll
  for (int j = 0; j < 4; ++j) {
    lacc[wave][4 * lane + j]       = acc[j];
    lacc[wave][128 + 4 * lane + j] = acc[4 + j];
  }
  __syncthreads();

  const int t = threadIdx.x;       // thread t owns dim t now
  float s = 0.0f;
#pragma unroll
  for (int w = 0; w < 8; ++w) s += lacc[w][t];
  const float mean = s / fmaxf((float)cnt, 1.0f);

  // block-reduce sum(mean^2)
  float mm = mean * mean;
#pragma unroll
  for (int o = 16; o >= 1; o >>= 1) mm += __shfl_xor(mm, o, 32);
  if (lane == 0) sred[wave] = mm;
  __syncthreads();
  float r = (t < 8) ? sred[t] : 0.0f;
  if (wave == 0) {
#pragma unroll
    for (int o = 4; o >= 1; o >>= 1) r += __shfl_xor(r, o, 32);
  }
  __syncthreads();
  if (t == 0) sred[0] = r;
  __syncthreads();
  const float mr2 = sred[0];
  const float mrn = sqrtf(mr2);

  const float cent = mean / fmaxf(mrn, 1e-12f);
  centroid_ws[(size_t)b * D + t] = cent;

  if (t == 0) {
    // cohesion = 1 - <mean_raw, centroid> = 1 - mr2 / max(mrn, 1e-12)
    const float coh = 1.0f - mr2 / fmaxf(mrn, 1e-12f);
    cohl_ws[b] = fmaxf(coh - tcoh[b], 0.0f);
  }
}

// ---------------------------------------------------------------- WMMA epilogue
// Per wave: 16 segments. Gram tile D = C * T^T over K=256 via v_wmma_f32_16x16x4_f32,
// diagonal gives <centroid_b, teacher_b>.
__global__ __launch_bounds__(32)
void wmma_loss_kernel(const float* __restrict__ cent,
                      const float* __restrict__ teacher,
                      const float* __restrict__ cohl,
                      float* __restrict__ out, int B) {
  const int b0   = blockIdx.x * 16;
  const int l    = threadIdx.x;
  const int m    = l & 15;       // row of A / col of B for this lane
  const int h    = l >> 4;       // lane half selects K sub-pair
  const int koff = h * 2;

  int bm = b0 + m;
  if (bm > B - 1) bm = B - 1;    // clamp (keeps EXEC all-ones; extras never written)
  const float* crow = cent    + (size_t)bm * D;
  const float* trow = teacher + (size_t)bm * D;

  v8f c = {0.f, 0.f, 0.f, 0.f, 0.f, 0.f, 0.f, 0.f};
#pragma unroll 4
  for (int kb = 0; kb < D; kb += 4) {
    // A (16x4 f32): lane m, VGPR j -> K = kb + koff + j
    // B (4x16 f32): lane n==m, VGPR j -> K = kb + koff + j  (B = T^T, symmetric pattern)
    v2f a, bb;
    a[0]  = crow[kb + koff];     a[1]  = crow[kb + koff + 1];
    bb[0] = trow[kb + koff];     bb[1] = trow[kb + koff + 1];
    c = __builtin_amdgcn_wmma_f32_16x16x4_f32(false, a, false, bb, (short)0, c,
                                              false, false);
  }

  // Diagonal of 16x16 f32 D tile: element M sits in VGPR (M&7) at lane M + 16*(M>=8).
#pragma unroll
  for (int v = 0; v < 8; ++v) {
    const int mdiag = v + 8 * h;
    if (m == mdiag) {
      const int b = b0 + mdiag;
      if (b < B) {
        const float closs = 1.0f - c[v];
        atomicAdd(out, (closs + cohl[b]) * (1.0f / (float)B));
      }
    }
  }
}

// ---------------------------------------------------------------- launch
extern "C" void kernel_launch(void* const* d_in, const int* in_sizes, int n_in,
                              void* d_out, int out_size, void* d_ws, size_t ws_size,
                              hipStream_t stream) {
  const float* emb     = (const float*)d_in[0];  // [N, 256]
  const float* teacher = (const float*)d_in[1];  // [B, 256]
  const float* tcoh    = (const float*)d_in[2];  // [B]
  const int*   midx    = (const int*)d_in[3];    // [M]
  const int*   segids  = (const int*)d_in[4];    // [M] sorted
  const int B = in_sizes[2];
  const int M = in_sizes[3];

  char*  ws     = (char*)d_ws;
  int*   counts = (int*)ws;
  int*   starts = counts + B;
  float* cohl   = (float*)(starts + B);
  float* cent   = cohl + B;                      // [B * 256]
  float* out    = (float*)d_out;

  const int zt = (B > out_size) ? B : out_size;
  zero_kernel<<<(zt + 255) / 256, 256, 0, stream>>>(counts, out, B, out_size);
  count_kernel<<<(M + 255) / 256, 256, 0, stream>>>(segids, counts, M);
  scan_kernel<<<1, 1024, 0, stream>>>(counts, starts, B);
  segment_kernel<<<B, 256, 0, stream>>>(emb, midx, starts, counts, tcoh, cent, cohl);
  wmma_loss_kernel<<<(B + 15) / 16, 32, 0, stream>>>(cent, teacher, cohl, out, B);
}